// GINConvNet_63685775065466
// MI455X (gfx1250) — compile-verified
//
#include <hip/hip_runtime.h>

typedef __attribute__((ext_vector_type(2))) float v2f;
typedef __attribute__((ext_vector_type(8))) float v8f;

#define N_NODES 100000
#define N_EDGES 3200000
#define DIM     32
#define IN_DIM  78
#define CHUNKS  (N_NODES / 16)   // 6250 exactly

__device__ __forceinline__ v8f wmma_f32(v2f a, v2f b, v8f c) {
    // V_WMMA_F32_16X16X4_F32 : D(16x16 f32) = A(16x4 f32) * B(4x16 f32) + C
    return __builtin_amdgcn_wmma_f32_16x16x4_f32(
        /*neg_a=*/false, a, /*neg_b=*/false, b,
        /*c_mod=*/(short)0, c, /*reuse_a=*/false, /*reuse_b=*/false);
}

// Guaranteed-native fp32 atomic add (no CAS-loop fallback): VGLOBAL
// GLOBAL_ATOMIC_ADD_F32, no-return form (tracked with STOREcnt).
__device__ __forceinline__ void atomic_add_f32_native(float* p, float v) {
    asm volatile("global_atomic_add_f32 %0, %1, off"
                 :
                 : "v"(p), "v"(v)
                 : "memory");
}

// ---------------------------------------------------------------------------
// Layer-1 projection: p = x @ W1a   (100000x78 @ 78x32), K padded 78->80.
// Writes p to BOTH h and z so the edge-scatter can accumulate on top of z.
// ---------------------------------------------------------------------------
__global__ __launch_bounds__(256) void proj_kernel(
    const float* __restrict__ x, const float* __restrict__ w1a,
    float* __restrict__ h, float* __restrict__ z) {
    __shared__ float Ws[80 * DIM];
    for (int i = threadIdx.x; i < 80 * DIM; i += blockDim.x) {
        int r = i >> 5;
        Ws[i] = (r < IN_DIM) ? w1a[i] : 0.0f;
    }
    __syncthreads();

    const int w = threadIdx.x >> 5, lane = threadIdx.x & 31;
    const int m = lane & 15, kh = lane >> 4;

    for (int chunk = blockIdx.x * 8 + w; chunk < CHUNKS; chunk += gridDim.x * 8) {
        const int base = chunk * 16;
        v8f acc0 = {}, acc1 = {};
        #pragma unroll
        for (int k0 = 0; k0 < 80; k0 += 4) {
            const int k = k0 + 2 * kh;
            const long rowoff = (long)(base + m) * IN_DIM;
            float ax = (k     < IN_DIM) ? x[rowoff + k]     : 0.0f;
            float ay = (k + 1 < IN_DIM) ? x[rowoff + k + 1] : 0.0f;
            v2f a  = {ax, ay};
            v2f b0 = {Ws[k * DIM + m],      Ws[(k + 1) * DIM + m]};
            v2f b1 = {Ws[k * DIM + 16 + m], Ws[(k + 1) * DIM + 16 + m]};
            acc0 = wmma_f32(a, b0, acc0);
            acc1 = wmma_f32(a, b1, acc1);
        }
        #pragma unroll
        for (int r = 0; r < 8; ++r) {
            const int row = base + r + 8 * kh;
            h[row * DIM + m]      = acc0[r];
            z[row * DIM + m]      = acc0[r];
            h[row * DIM + 16 + m] = acc1[r];
            z[row * DIM + 16 + m] = acc1[r];
        }
    }
}

// ---------------------------------------------------------------------------
// Edge scatter: z[dst] += h[src].  One wave per edge; lane == feature dim.
// Coalesced 128B gather + native 128B f32-atomic burst per edge.
// ---------------------------------------------------------------------------
__global__ __launch_bounds__(256) void edge_scatter(
    const int* __restrict__ ei, const float* __restrict__ h, float* z) {
    const int lane   = threadIdx.x & 31;
    const int warp   = (blockIdx.x * blockDim.x + threadIdx.x) >> 5;
    const int nwarps = (gridDim.x * blockDim.x) >> 5;
    for (int e = warp; e < N_EDGES; e += nwarps) {
        const int src = ei[e];
        const int dst = ei[N_EDGES + e];
        const float v = h[src * DIM + lane];
        atomic_add_f32_native(&z[dst * DIM + lane], v);
    }
}

// ---------------------------------------------------------------------------
// GIN MLP: h = relu( relu(z@Wa + Ba) @ Wb + Bb )   (skip_first: z already =
// agg of projected features, so first stage is just relu(z + Ba)).
// One wave per 16-node chunk; relayout C->A via per-wave LDS tile.
// ---------------------------------------------------------------------------
__global__ __launch_bounds__(256) void gin_mlp(
    const float* zin,
    const float* __restrict__ Wa, const float* __restrict__ Ba,
    const float* __restrict__ Wb, const float* __restrict__ Bb,
    float* h, float* zout, int skip_first, int write_z) {
    __shared__ float WaS[DIM * DIM], WbS[DIM * DIM], BaS[DIM], BbS[DIM];
    __shared__ float y1[8][16 * DIM];

    for (int i = threadIdx.x; i < DIM * DIM; i += blockDim.x) {
        WaS[i] = skip_first ? 0.0f : Wa[i];
        WbS[i] = Wb[i];
    }
    if (threadIdx.x < DIM) {
        BaS[threadIdx.x] = Ba[threadIdx.x];
        BbS[threadIdx.x] = Bb[threadIdx.x];
    }
    __syncthreads();

    const int w = threadIdx.x >> 5, lane = threadIdx.x & 31;
    const int m = lane & 15, kh = lane >> 4;
    float* Y = y1[w];

    for (int chunk = blockIdx.x * 8 + w; chunk < CHUNKS; chunk += gridDim.x * 8) {
        const int base = chunk * 16;

        if (skip_first) {
            #pragma unroll
            for (int r = 0; r < 16; ++r) {
                const float v = zin[(base + r) * DIM + lane] + BaS[lane];
                Y[r * DIM + lane] = fmaxf(v, 0.0f);
            }
        } else {
            v8f acc0 = {}, acc1 = {};
            #pragma unroll
            for (int k0 = 0; k0 < DIM; k0 += 4) {
                const int k = k0 + 2 * kh;
                const float* zr = &zin[(base + m) * DIM + k];
                v2f a  = {zr[0], zr[1]};
                v2f b0 = {WaS[k * DIM + m],      WaS[(k + 1) * DIM + m]};
                v2f b1 = {WaS[k * DIM + 16 + m], WaS[(k + 1) * DIM + 16 + m]};
                acc0 = wmma_f32(a, b0, acc0);
                acc1 = wmma_f32(a, b1, acc1);
            }
            #pragma unroll
            for (int r = 0; r < 8; ++r) {
                const int row = r + 8 * kh;
                Y[row * DIM + m]      = fmaxf(acc0[r] + BaS[m],      0.0f);
                Y[row * DIM + 16 + m] = fmaxf(acc1[r] + BaS[16 + m], 0.0f);
            }
        }
        // cross-lane LDS dependency within one lockstepped wave32:
        asm volatile("s_wait_dscnt 0" ::: "memory");

        v8f acc0 = {}, acc1 = {};
        #pragma unroll
        for (int k0 = 0; k0 < DIM; k0 += 4) {
            const int k = k0 + 2 * kh;
            v2f a  = {Y[m * DIM + k], Y[m * DIM + k + 1]};
            v2f b0 = {WbS[k * DIM + m],      WbS[(k + 1) * DIM + m]};
            v2f b1 = {WbS[k * DIM + 16 + m], WbS[(k + 1) * DIM + 16 + m]};
            acc0 = wmma_f32(a, b0, acc0);
            acc1 = wmma_f32(a, b1, acc1);
        }
        #pragma unroll
        for (int r = 0; r < 8; ++r) {
            const int row = base + r + 8 * kh;
            const float v0 = fmaxf(acc0[r] + BbS[m],      0.0f);
            const float v1 = fmaxf(acc1[r] + BbS[16 + m], 0.0f);
            h[row * DIM + m]      = v0;
            h[row * DIM + 16 + m] = v1;
            if (write_z) {  // seed z = h for the next layer's scatter-add
                zout[row * DIM + m]      = v0;
                zout[row * DIM + 16 + m] = v1;
            }
        }
    }
}

// ---------------------------------------------------------------------------
// Global add-pool: per-block partial sums of h over nodes (lane == dim).
// ---------------------------------------------------------------------------
__global__ __launch_bounds__(256) void pool_partial(
    const float* __restrict__ h, float* __restrict__ partials) {
    __shared__ float red[8][DIM];
    const int w = threadIdx.x >> 5, lane = threadIdx.x & 31;
    const int gw = (blockIdx.x * blockDim.x + threadIdx.x) >> 5;
    const int nw = (gridDim.x * blockDim.x) >> 5;
    float acc = 0.0f;
    for (int row = gw; row < N_NODES; row += nw) acc += h[row * DIM + lane];
    red[w][lane] = acc;
    __syncthreads();
    if (w == 0) {
        float s = 0.0f;
        #pragma unroll
        for (int i = 0; i < 8; ++i) s += red[i][lane];
        partials[blockIdx.x * DIM + lane] = s;
    }
}

// ---------------------------------------------------------------------------
// Head: g = sum(partials); out = relu(g@fc_w + fc_b) @ out_w + out_b.
// Single wave32; lane == output column.
// ---------------------------------------------------------------------------
__global__ void final_head(const float* __restrict__ partials, int nparts,
                           const float* __restrict__ fc_w, const float* __restrict__ fc_b,
                           const float* __restrict__ out_w, const float* __restrict__ out_b,
                           float* __restrict__ out) {
    __shared__ float gS[DIM], tS[DIM];
    const int lane = threadIdx.x;  // 0..31
    float g = 0.0f;
    for (int b = 0; b < nparts; ++b) g += partials[b * DIM + lane];
    gS[lane] = g;
    asm volatile("s_wait_dscnt 0" ::: "memory");
    float t = fc_b[lane];
    #pragma unroll
    for (int k = 0; k < DIM; ++k) t += gS[k] * fc_w[k * DIM + lane];
    tS[lane] = fmaxf(t, 0.0f);
    asm volatile("s_wait_dscnt 0" ::: "memory");
    float o = out_b[lane];
    #pragma unroll
    for (int k = 0; k < DIM; ++k) o += tS[k] * out_w[k * DIM + lane];
    out[lane] = o;
}

// ---------------------------------------------------------------------------
extern "C" void kernel_launch(void* const* d_in, const int* in_sizes, int n_in,
                              void* d_out, int out_size, void* d_ws, size_t ws_size,
                              hipStream_t stream) {
    (void)in_sizes; (void)n_in; (void)out_size; (void)ws_size;

    const float* x    = (const float*)d_in[0];
    const int*   ei   = (const int*)  d_in[1];
    const float* w1a  = (const float*)d_in[2];
    const float* b1a  = (const float*)d_in[3];
    const float* w1b  = (const float*)d_in[4];
    const float* b1b  = (const float*)d_in[5];
    const float* wa   = (const float*)d_in[6];
    const float* ba   = (const float*)d_in[7];
    const float* wb   = (const float*)d_in[8];
    const float* bb   = (const float*)d_in[9];
    const float* fc_w = (const float*)d_in[10];
    const float* fc_b = (const float*)d_in[11];
    const float* out_w= (const float*)d_in[12];
    const float* out_b= (const float*)d_in[13];
    float* out = (float*)d_out;

    char* ws = (char*)d_ws;
    float* z        = (float*)ws;                        // 100000*32 f32 = 12.8 MB
    float* h        = (float*)(ws + 12800000);           // 100000*32 f32 = 12.8 MB
    float* partials = (float*)(ws + 25600000);           // 240*32 f32

    const dim3 blk(256);
    const int mlp_blocks  = (CHUNKS + 7) / 8;            // 782: one wave per 16-node chunk
    const int edge_blocks = 4096;                        // ~98 edges per wave, grid-stride
    const int pool_blocks = 240;

    // Layer 1 with project-then-aggregate: p = x@W1a ; z = p + scatter(p)
    proj_kernel<<<mlp_blocks, blk, 0, stream>>>(x, w1a, h, z);
    edge_scatter<<<edge_blocks, blk, 0, stream>>>(ei, h, z);
    gin_mlp<<<mlp_blocks, blk, 0, stream>>>(z, w1a, b1a, w1b, b1b, h, z,
                                            /*skip_first=*/1, /*write_z=*/1);
    // Layers 2..5
    for (int i = 0; i < 4; ++i) {
        edge_scatter<<<edge_blocks, blk, 0, stream>>>(ei, h, z);
        gin_mlp<<<mlp_blocks, blk, 0, stream>>>(z, wa + i * DIM * DIM, ba + i * DIM,
                                                wb + i * DIM * DIM, bb + i * DIM,
                                                h, z, /*skip_first=*/0,
                                                /*write_z=*/(i < 3) ? 1 : 0);
    }
    // Pool + dense head
    pool_partial<<<pool_blocks, blk, 0, stream>>>(h, partials);
    final_head<<<1, 32, 0, stream>>>(partials, pool_blocks, fc_w, fc_b, out_w, out_b, out);
}